// SGCN_14302241095851
// MI455X (gfx1250) — compile-verified
//
#include <hip/hip_runtime.h>
#include <math.h>

// ---------------------------------------------------------------------------
// SGCN 2-layer GCN for MI455X (gfx1250), wave32, WMMA f32 16x16x4.
// Problem: N=100000 nodes, E=1.6M edges, d=128, h=64, c=10.
// Strategy: FP32 WMMA for both GEMMs; edge gather/scatter via f32 global
// atomics that stay resident in the 192MB L2 (working set ~58MB).
// ---------------------------------------------------------------------------

typedef __attribute__((ext_vector_type(2))) float v2f;
typedef __attribute__((ext_vector_type(8))) float v8f;

#define D_IN   128
#define D_HID  64
#define D_OUT  10
#define D_OUTP 16   // padded class dim

// ---------------- degree / normalization ----------------

__global__ void k_deg_init(float* __restrict__ deg, int N) {
    int i = blockIdx.x * blockDim.x + threadIdx.x;
    if (i < N) deg[i] = 1.0f;  // self loop
}

__global__ void k_deg_edges(const int* __restrict__ dst, float* __restrict__ deg, int E) {
    int e = blockIdx.x * blockDim.x + threadIdx.x;
    if (e < E) atomicAdd(&deg[dst[e]], 1.0f);
}

__global__ void k_rsqrt(float* __restrict__ dinv, int N) {
    int i = blockIdx.x * blockDim.x + threadIdx.x;
    if (i < N) dinv[i] = rsqrtf(dinv[i]);  // deg >= 1 always (self loop)
}

// ---------------- GEMM1: H1 = x @ W1  [N,128]@[128,64] ----------------
// One wave computes a 16x16 output tile with V_WMMA_F32_16X16X4_F32.
// A-layout (16x4 f32): lanes 0-15 row M=lane, K={k,k+1}; lanes 16-31 K={k+2,k+3}.
// B-layout (4x16 f32): lanes = N col; VGPR0/1 hold K rows {k,k+1} (lo half) /
// {k+2,k+3} (hi half). D: VGPR v -> row v (lanes 0-15), row v+8 (lanes 16-31).

__global__ void __launch_bounds__(256)
k_gemm1(const float* __restrict__ x, const float* __restrict__ W,
        float* __restrict__ H, int N) {
    int wid  = (blockIdx.x * blockDim.x + threadIdx.x) >> 5;  // uniform per wave
    int lane = threadIdx.x & 31;
    int ntM  = (N + 15) >> 4;
    if (wid >= ntM * (D_HID / 16)) return;  // whole-wave uniform exit
    int tM = wid >> 2;
    int tN = wid & 3;

    int l16  = lane & 15;
    int half = (lane >= 16) ? 2 : 0;
    int row  = tM * 16 + l16; if (row >= N) row = N - 1;  // clamp, keep EXEC full
    int col  = tN * 16 + l16;

    const float* xr = x + (size_t)row * D_IN;
    v8f c = {0.f,0.f,0.f,0.f,0.f,0.f,0.f,0.f};
    #pragma unroll 4
    for (int k = 0; k < D_IN; k += 4) {
        v2f a, b;
        a.x = xr[k + half];
        a.y = xr[k + half + 1];
        b.x = W[(size_t)(k + half) * D_HID + col];
        b.y = W[(size_t)(k + half + 1) * D_HID + col];
        c = __builtin_amdgcn_wmma_f32_16x16x4_f32(false, a, false, b,
                                                  (short)0, c, false, false);
    }
    int rbase = tM * 16 + ((lane >= 16) ? 8 : 0);
    #pragma unroll
    for (int v = 0; v < 8; ++v) {
        int r = rbase + v;
        if (r < N) H[(size_t)r * D_HID + tN * 16 + l16] = c[v];
    }
}

// ---------------- layer-1 aggregation ----------------

// agg[i,f] = b1[f] + dinv[i]^2 * H1[i,f]   (self-loop message + bias)
__global__ void k_agg1_init(const float* __restrict__ H1, const float* __restrict__ dinv,
                            const float* __restrict__ b1, float* __restrict__ agg, int N) {
    long long gid = (long long)blockIdx.x * blockDim.x + threadIdx.x;
    int i = (int)(gid >> 6);
    int f = (int)(gid & 63);
    if (i >= N) return;
    float di = dinv[i];
    agg[(size_t)i * D_HID + f] = b1[f] + di * di * H1[(size_t)i * D_HID + f];
}

// one wave per edge: 32 lanes x float2 = 64 features
__global__ void k_scatter1(const float* __restrict__ H1, const float* __restrict__ dinv,
                           const int* __restrict__ src, const int* __restrict__ dst,
                           float* __restrict__ agg, int E) {
    long long gid = (long long)blockIdx.x * blockDim.x + threadIdx.x;
    int e    = (int)(gid >> 5);
    int lane = (int)(gid & 31);
    if (e >= E) return;
    int s = src[e], d = dst[e];
    float norm = dinv[s] * dinv[d];
    const float2* hp = (const float2*)(H1 + (size_t)s * D_HID);
    float2 h = hp[lane];
    float* ap = agg + (size_t)d * D_HID + lane * 2;
    atomicAdd(ap,     h.x * norm);
    atomicAdd(ap + 1, h.y * norm);
}

// ---------------- GEMM2: H2 = relu(agg) @ W2  [N,64]@[64,10->16] ----------------

__global__ void __launch_bounds__(256)
k_gemm2(const float* __restrict__ A, const float* __restrict__ W2,
        float* __restrict__ H2, int N) {
    int wid  = (blockIdx.x * blockDim.x + threadIdx.x) >> 5;
    int lane = threadIdx.x & 31;
    int ntM  = (N + 15) >> 4;
    if (wid >= ntM) return;

    int l16  = lane & 15;
    int half = (lane >= 16) ? 2 : 0;
    int row  = wid * 16 + l16; if (row >= N) row = N - 1;
    int col  = l16;                 // cols 10..15 are zero-padded
    bool cv  = (col < D_OUT);

    const float* ar = A + (size_t)row * D_HID;
    v8f c = {0.f,0.f,0.f,0.f,0.f,0.f,0.f,0.f};
    #pragma unroll 4
    for (int k = 0; k < D_HID; k += 4) {
        v2f a, b;
        a.x = fmaxf(ar[k + half], 0.0f);          // fused ReLU
        a.y = fmaxf(ar[k + half + 1], 0.0f);
        b.x = cv ? W2[(size_t)(k + half) * D_OUT + col] : 0.0f;
        b.y = cv ? W2[(size_t)(k + half + 1) * D_OUT + col] : 0.0f;
        c = __builtin_amdgcn_wmma_f32_16x16x4_f32(false, a, false, b,
                                                  (short)0, c, false, false);
    }
    int rbase = wid * 16 + ((lane >= 16) ? 8 : 0);
    #pragma unroll
    for (int v = 0; v < 8; ++v) {
        int r = rbase + v;
        if (r < N) H2[(size_t)r * D_OUTP + l16] = c[v];
    }
}

// ---------------- layer-2 aggregation ----------------

__global__ void k_agg2_init(const float* __restrict__ H2, const float* __restrict__ dinv,
                            const float* __restrict__ b2, float* __restrict__ logits, int N) {
    long long gid = (long long)blockIdx.x * blockDim.x + threadIdx.x;
    int i = (int)(gid >> 4);
    int f = (int)(gid & 15);
    if (i >= N) return;
    float v = 0.0f;
    if (f < D_OUT) {
        float di = dinv[i];
        v = b2[f] + di * di * H2[(size_t)i * D_OUTP + f];
    }
    logits[(size_t)i * D_OUTP + f] = v;
}

// 16 threads per edge, 10 active classes
__global__ void k_scatter2(const float* __restrict__ H2, const float* __restrict__ dinv,
                           const int* __restrict__ src, const int* __restrict__ dst,
                           float* __restrict__ logits, int E) {
    long long gid = (long long)blockIdx.x * blockDim.x + threadIdx.x;
    int e = (int)(gid >> 4);
    int f = (int)(gid & 15);
    if (e >= E || f >= D_OUT) return;
    int s = src[e], d = dst[e];
    float norm = dinv[s] * dinv[d];
    atomicAdd(&logits[(size_t)d * D_OUTP + f], H2[(size_t)s * D_OUTP + f] * norm);
}

// ---------------- evidential softmax ----------------

__global__ void k_final(const float* __restrict__ logits, float* __restrict__ out, int N) {
    int i = blockIdx.x * blockDim.x + threadIdx.x;
    if (i >= N) return;
    float a[D_OUT];
    float s = 0.0f;
    #pragma unroll
    for (int f = 0; f < D_OUT; ++f) {
        a[f] = 1.0f + expf(logits[(size_t)i * D_OUTP + f]);
        s += a[f];
    }
    float inv = 1.0f / s;
    #pragma unroll
    for (int f = 0; f < D_OUT; ++f)
        out[(size_t)i * D_OUT + f] = a[f] * inv;
}

// ---------------------------------------------------------------------------

extern "C" void kernel_launch(void* const* d_in, const int* in_sizes, int n_in,
                              void* d_out, int out_size, void* d_ws, size_t ws_size,
                              hipStream_t stream) {
    (void)n_in; (void)out_size; (void)ws_size;

    const float* x   = (const float*)d_in[0];
    const int*   ei  = (const int*)d_in[1];   // [2, E] flat
    const float* W1  = (const float*)d_in[2];
    const float* b1  = (const float*)d_in[3];
    const float* W2  = (const float*)d_in[4];
    const float* b2  = (const float*)d_in[5];
    float*       out = (float*)d_out;

    const int N = in_sizes[0] / D_IN;
    const int E = in_sizes[1] / 2;
    const int* src = ei;
    const int* dst = ei + E;

    // workspace layout (256B aligned slabs)
    char* ws = (char*)d_ws;
    size_t off = 0;
    auto alloc = [&](size_t bytes) -> char* {
        char* p = ws + off;
        off += (bytes + 255) & ~(size_t)255;
        return p;
    };
    float* dinv   = (float*)alloc((size_t)N * sizeof(float));            //  0.4 MB
    float* H1     = (float*)alloc((size_t)N * D_HID * sizeof(float));    // 25.6 MB
    float* agg1   = (float*)alloc((size_t)N * D_HID * sizeof(float));    // 25.6 MB
    float* H2     = (float*)alloc((size_t)N * D_OUTP * sizeof(float));   //  6.4 MB
    float* logits = (float*)alloc((size_t)N * D_OUTP * sizeof(float));   //  6.4 MB

    const int B = 256;
    auto cdiv = [](long long a, long long b) { return (int)((a + b - 1) / b); };

    // degrees -> dinv
    k_deg_init <<<cdiv(N, B), B, 0, stream>>>(dinv, N);
    k_deg_edges<<<cdiv(E, B), B, 0, stream>>>(dst, dinv, E);
    k_rsqrt    <<<cdiv(N, B), B, 0, stream>>>(dinv, N);

    // layer 1
    long long waves1 = (long long)((N + 15) / 16) * (D_HID / 16);
    k_gemm1    <<<cdiv(waves1 * 32, B), B, 0, stream>>>(x, W1, H1, N);
    k_agg1_init<<<cdiv((long long)N * D_HID, B), B, 0, stream>>>(H1, dinv, b1, agg1, N);
    k_scatter1 <<<cdiv((long long)E * 32, B), B, 0, stream>>>(H1, dinv, src, dst, agg1, E);

    // layer 2 (ReLU fused into GEMM2 A-load)
    long long waves2 = (long long)((N + 15) / 16);
    k_gemm2    <<<cdiv(waves2 * 32, B), B, 0, stream>>>(agg1, W2, H2, N);
    k_agg2_init<<<cdiv((long long)N * D_OUTP, B), B, 0, stream>>>(H2, dinv, b2, logits, N);
    k_scatter2 <<<cdiv((long long)E * D_OUTP, B), B, 0, stream>>>(H2, dinv, src, dst, logits, E);

    // evidential softmax
    k_final    <<<cdiv(N, B), B, 0, stream>>>(logits, out, N);
}